// SpatialWeightingV2_44006234915180
// MI455X (gfx1250) — compile-verified
//
#include <hip/hip_runtime.h>
#include <math.h>

// Problem constants (match reference)
#define BATCH 16
#define CH    512
#define HW    4096    // 64*64
#define ICH   32      // CH/16

typedef float v2f __attribute__((ext_vector_type(2)));
typedef float v8f __attribute__((ext_vector_type(8)));

static __device__ __forceinline__ float sigmoidf_(float v) {
    return 1.0f / (1.0f + __expf(-v));
}

// ---------------------------------------------------------------------------
// K1: xmean[b*512+c] = mean over HW of x[b,c,:]
// ---------------------------------------------------------------------------
__global__ void row_mean_kernel(const float* __restrict__ x, float* __restrict__ xmean) {
    __shared__ float red[256];
    const int row = blockIdx.x;                       // b*CH + c
    const float* p = x + (size_t)row * HW;
    float s = 0.0f;
#pragma unroll 4
    for (int n = threadIdx.x; n < HW; n += 256) s += p[n];
    red[threadIdx.x] = s;
    __syncthreads();
    for (int off = 128; off > 0; off >>= 1) {
        if (threadIdx.x < off) red[threadIdx.x] += red[threadIdx.x + off];
        __syncthreads();
    }
    if (threadIdx.x == 0) xmean[row] = red[0] * (1.0f / (float)HW);
}

// ---------------------------------------------------------------------------
// K2: qlog[b*HW+n] = sum_c w_q_channel[c] * x[b,c,n]   (coalesced over n)
// ---------------------------------------------------------------------------
__global__ void q_logits_kernel(const float* __restrict__ x,
                                const float* __restrict__ wq,
                                float* __restrict__ qlog) {
    const int b = blockIdx.y;
    const int n = blockIdx.x * 256 + threadIdx.x;
    const float* xb = x + (size_t)b * CH * HW + n;
    float acc = 0.0f;
#pragma unroll 8
    for (int c = 0; c < CH; ++c) acc += wq[c] * xb[(size_t)c * HW];
    qlog[b * HW + n] = acc;
}

// ---------------------------------------------------------------------------
// K3: in-place softmax over HW for each batch
// ---------------------------------------------------------------------------
__global__ void spatial_softmax_kernel(float* __restrict__ q) {
    __shared__ float red[256];
    float* p = q + (size_t)blockIdx.x * HW;
    const int t = threadIdx.x;
    float mx = -3.0e38f;
    for (int n = t; n < HW; n += 256) mx = fmaxf(mx, p[n]);
    red[t] = mx; __syncthreads();
    for (int off = 128; off > 0; off >>= 1) {
        if (t < off) red[t] = fmaxf(red[t], red[t + off]);
        __syncthreads();
    }
    mx = red[0]; __syncthreads();
    float s = 0.0f;
    for (int n = t; n < HW; n += 256) s += __expf(p[n] - mx);
    red[t] = s; __syncthreads();
    for (int off = 128; off > 0; off >>= 1) {
        if (t < off) red[t] += red[t + off];
        __syncthreads();
    }
    const float inv = 1.0f / red[0];
    for (int n = t; n < HW; n += 256) p[n] = __expf(p[n] - mx) * inv;
}

// ---------------------------------------------------------------------------
// K4: xq[b*512+c] = sum_n x[b,c,n] * q[b,n]   (weighted spatial pooling)
// ---------------------------------------------------------------------------
__global__ void xq_kernel(const float* __restrict__ x,
                          const float* __restrict__ q,
                          float* __restrict__ xq) {
    __shared__ float red[256];
    const int row = blockIdx.x;                       // b*CH + c
    const int b = row >> 9;                           // /CH
    const float* p  = x + (size_t)row * HW;
    const float* qb = q + (size_t)b * HW;
    float s = 0.0f;
#pragma unroll 4
    for (int n = threadIdx.x; n < HW; n += 256) s += p[n] * qb[n];
    red[threadIdx.x] = s;
    __syncthreads();
    for (int off = 128; off > 0; off >>= 1) {
        if (threadIdx.x < off) red[threadIdx.x] += red[threadIdx.x + off];
        __syncthreads();
    }
    if (threadIdx.x == 0) xq[row] = red[0];
}

// ---------------------------------------------------------------------------
// K5a: WMMA small GEMMs, D[m,n] = sum_k A[m,k]*B_src[n,k]
//   sel=0: y[i,b]      = w_v_channel[i,:] . xq[b,:]      (M=32,K=512,N=16)
//   sel=1: qs_raw[i,b] = w_q_spatial[i,:] . xmean[b,:]   (M=32,K=512,N=16)
// One block, 4 waves: wave = sel*2 + mtile. Uses V_WMMA_F32_16X16X4_F32.
// A frag (16x4): lane L: m=L&15, khalf=L>>4; vgpr0=A[m][k0+2*khalf], vgpr1=+1
// B frag (4x16): lane L: n=L&15, khalf=L>>4; vgpr0=B[k0+2*khalf][n], vgpr1=+1
// D (16x16): vgpr g: row = g + 8*(L>>4), col = L&15
// ---------------------------------------------------------------------------
__global__ void wmma_y_qs_kernel(const float* __restrict__ w_v_channel,
                                 const float* __restrict__ w_q_spatial,
                                 const float* __restrict__ xq,     // [B][CH]
                                 const float* __restrict__ xmean,  // [B][CH]
                                 float* __restrict__ y_ws,         // [ICH][B]
                                 float* __restrict__ qsraw_ws) {   // [ICH][B]
    const int wave  = threadIdx.x >> 5;
    const int lane  = threadIdx.x & 31;
    const int sel   = wave >> 1;
    const int mtile = wave & 1;
    const float* A  = sel ? w_q_spatial : w_v_channel;  // [ICH][CH]
    const float* Bm = sel ? xmean       : xq;           // [B][CH]; B[k][n]=Bm[n*CH+k]
    float* Dst      = sel ? qsraw_ws    : y_ws;

    const int m     = lane & 15;
    const int khalf = lane >> 4;
    const float* Arow = A  + (size_t)(mtile * 16 + m) * CH;
    const float* Brow = Bm + (size_t)m * CH;            // n == lane&15 == m

    v8f acc = {};
    for (int k0 = 0; k0 < CH; k0 += 4) {
        const int k = k0 + 2 * khalf;
        v2f a, bf;
        a.x  = Arow[k];
        a.y  = Arow[k + 1];
        bf.x = Brow[k];
        bf.y = Brow[k + 1];
        acc = __builtin_amdgcn_wmma_f32_16x16x4_f32(false, a, false, bf,
                                                    (short)0, acc, false, false);
    }
#pragma unroll
    for (int g = 0; g < 8; ++g) {
        const int row = mtile * 16 + g + 8 * khalf;     // i index
        Dst[row * 16 + (lane & 15)] = acc[g];
    }
}

// ---------------------------------------------------------------------------
// K5b: softmax over internal channels (IC=32) per batch
// ---------------------------------------------------------------------------
__global__ void qs_softmax_kernel(const float* __restrict__ qsraw,  // [ICH][B]
                                  float* __restrict__ qs) {         // [ICH][B]
    const int b = threadIdx.x;
    if (b < BATCH) {
        float mx = -3.0e38f;
        for (int i = 0; i < ICH; ++i) mx = fmaxf(mx, qsraw[i * 16 + b]);
        float s = 0.0f;
        for (int i = 0; i < ICH; ++i) s += __expf(qsraw[i * 16 + b] - mx);
        const float inv = 1.0f / s;
        for (int i = 0; i < ICH; ++i)
            qs[i * 16 + b] = __expf(qsraw[i * 16 + b] - mx) * inv;
    }
}

// ---------------------------------------------------------------------------
// K5c: WMMA small GEMMs over K=32:
//   wave0: scale[b,c]  = sigmoid( sum_i w_out[c,i]*y[i,b] + b_out[c] )
//   wave1: wq_eff[b,c] = sum_i w_v_spatial[i,c]*qs[i,b]
// grid = 32 c-tiles of 16; block = 64 threads (2 waves, EXEC all-ones each).
// ---------------------------------------------------------------------------
__global__ void wmma_scale_wqeff_kernel(const float* __restrict__ w_out,       // [CH][ICH]
                                        const float* __restrict__ b_out,       // [CH]
                                        const float* __restrict__ w_v_spatial, // [ICH][CH]
                                        const float* __restrict__ y_ws,        // [ICH][B]
                                        const float* __restrict__ qs_ws,       // [ICH][B]
                                        float* __restrict__ scale_ws,          // [B][CH]
                                        float* __restrict__ wqeff_ws) {        // [B][CH]
    const int wave  = threadIdx.x >> 5;
    const int lane  = threadIdx.x & 31;
    const int c0    = blockIdx.x * 16;
    const int m     = lane & 15;
    const int khalf = lane >> 4;

    v8f acc = {};
    for (int k0 = 0; k0 < ICH; k0 += 4) {
        const int k = k0 + 2 * khalf;
        v2f a, bf;
        if (wave == 0) {
            a.x  = w_out[(size_t)(c0 + m) * ICH + k];
            a.y  = w_out[(size_t)(c0 + m) * ICH + k + 1];
            bf.x = y_ws[k * 16 + m];
            bf.y = y_ws[(k + 1) * 16 + m];
        } else {
            a.x  = w_v_spatial[(size_t)k * CH + (c0 + m)];
            a.y  = w_v_spatial[(size_t)(k + 1) * CH + (c0 + m)];
            bf.x = qs_ws[k * 16 + m];
            bf.y = qs_ws[(k + 1) * 16 + m];
        }
        acc = __builtin_amdgcn_wmma_f32_16x16x4_f32(false, a, false, bf,
                                                    (short)0, acc, false, false);
    }
#pragma unroll
    for (int g = 0; g < 8; ++g) {
        const int c = c0 + g + 8 * khalf;
        const int b = lane & 15;
        if (wave == 0) {
            scale_ws[b * CH + c] = sigmoidf_(acc[g] + b_out[c]);
        } else {
            wqeff_ws[b * CH + c] = acc[g];
        }
    }
}

// ---------------------------------------------------------------------------
// K6: fused spatial-attention GEMV + final output.
//   ys = sigmoid( sum_c wq_eff[b,c]*x[b,c,n] )
//   out[b,c,n] = x[b,c,n] * (scale[b,c] + ys)
// Second c-loop re-reads the same x tile (L2-resident: 134 MB x < 192 MB L2).
// ---------------------------------------------------------------------------
__global__ void output_kernel(const float* __restrict__ x,
                              const float* __restrict__ scale_ws,  // [B][CH]
                              const float* __restrict__ wqeff_ws,  // [B][CH]
                              float* __restrict__ out) {
    const int b = blockIdx.y;
    const int n = blockIdx.x * 256 + threadIdx.x;
    const float* xb = x + (size_t)b * CH * HW + n;
    const float* we = wqeff_ws + b * CH;
    const float* sc = scale_ws + b * CH;
    float* ob = out + (size_t)b * CH * HW + n;

    float acc = 0.0f;
#pragma unroll 8
    for (int c = 0; c < CH; ++c) acc += we[c] * xb[(size_t)c * HW];
    const float ys = sigmoidf_(acc);

#pragma unroll 8
    for (int c = 0; c < CH; ++c)
        ob[(size_t)c * HW] = xb[(size_t)c * HW] * (sc[c] + ys);
}

// ---------------------------------------------------------------------------
extern "C" void kernel_launch(void* const* d_in, const int* in_sizes, int n_in,
                              void* d_out, int out_size, void* d_ws, size_t ws_size,
                              hipStream_t stream) {
    (void)in_sizes; (void)n_in; (void)out_size; (void)ws_size;
    const float* x            = (const float*)d_in[0];
    const float* w_v_channel  = (const float*)d_in[1];
    const float* w_q_channel  = (const float*)d_in[2];
    const float* w_out        = (const float*)d_in[3];
    const float* b_out        = (const float*)d_in[4];
    const float* w_v_spatial  = (const float*)d_in[5];
    const float* w_q_spatial  = (const float*)d_in[6];
    float* out = (float*)d_out;

    char* ws = (char*)d_ws;
    float* qlog  = (float*)(ws);                                   // B*HW   = 65536
    float* xmean = (float*)(ws + (size_t)(65536)          * 4);    // B*CH   = 8192
    float* xq    = (float*)(ws + (size_t)(65536 + 8192)   * 4);    // B*CH   = 8192
    float* scale = (float*)(ws + (size_t)(65536 + 16384)  * 4);    // B*CH   = 8192
    float* wqeff = (float*)(ws + (size_t)(65536 + 24576)  * 4);    // B*CH   = 8192
    float* y_ws  = (float*)(ws + (size_t)(65536 + 32768)  * 4);    // ICH*B  = 512
    float* qsraw = (float*)(ws + (size_t)(65536 + 33280)  * 4);    // ICH*B  = 512
    float* qs    = (float*)(ws + (size_t)(65536 + 33792)  * 4);    // ICH*B  = 512

    // 1) per-(b,c) spatial mean
    row_mean_kernel<<<BATCH * CH, 256, 0, stream>>>(x, xmean);
    // 2) channel-attention query logits over spatial dim
    q_logits_kernel<<<dim3(HW / 256, BATCH), 256, 0, stream>>>(x, w_q_channel, qlog);
    // 3) softmax over HW per batch (in place)
    spatial_softmax_kernel<<<BATCH, 256, 0, stream>>>(qlog);
    // 4) q-weighted spatial pooling
    xq_kernel<<<BATCH * CH, 256, 0, stream>>>(x, qlog, xq);
    // 5) WMMA: y = Wvc . xq ; qs_raw = Wqs . xmean
    wmma_y_qs_kernel<<<1, 128, 0, stream>>>(w_v_channel, w_q_spatial, xq, xmean,
                                            y_ws, qsraw);
    // 6) softmax over internal channels
    qs_softmax_kernel<<<1, 32, 0, stream>>>(qsraw, qs);
    // 7) WMMA: scale = sigmoid(Wout.y + b) ; wq_eff = qs^T . Wvs
    wmma_scale_wqeff_kernel<<<CH / 16, 64, 0, stream>>>(w_out, b_out, w_v_spatial,
                                                        y_ws, qs, scale, wqeff);
    // 8) fused spatial GEMV + elementwise output
    output_kernel<<<dim3(HW / 256, BATCH), 256, 0, stream>>>(x, scale, wqeff, out);
}